// VaryingCoefficientsLayer_40295383171268
// MI455X (gfx1250) — compile-verified
//
#include <hip/hip_runtime.h>

// MI455X / gfx1250, wave32.
// D'[s,p] = nullspace^T x kernel^T via V_WMMA_F32_16X16X4_F32 (fp32, K=32 -> 8 WMMA/tile).
// A-tiles staged global->LDS with global_load_async_to_lds_b128, double-buffered
// (s_wait_asynccnt 4 overlaps next tile's copy with current tile's WMMAs).
typedef __attribute__((ext_vector_type(2))) float v2f;
typedef __attribute__((ext_vector_type(8))) float v8f;

#define XDIM 512
#define YDIM 512
#define KDIM 32   // IN_SIZE
#define SDIM 16   // 4x4 stencil

#define WAVES_PER_BLOCK 8
#define TILE 16                        // pixels per tile (WMMA N)
#define TILES_PER_WAVE 4               // 64 consecutive y per wave; block = one (b,x) row
#define LDS_ROW 36                     // 32 floats + 4 pad: 144B rows (16B aligned, 9m%16 bank spread)
#define LDS_TILE (TILE * LDS_ROW)      // floats per staged tile (2304 B)
#define LDS_WAVE (2 * LDS_TILE)        // double buffer per wave

__global__ __launch_bounds__(256)
void varcoef_wmma_kernel(const float* __restrict__ kern,      // (B,X,Y,32)
                         const float* __restrict__ source,    // (B,X,Y)
                         const float* __restrict__ nullspace, // (32,16)
                         const float* __restrict__ bias,      // (16,)
                         float* __restrict__ out)             // (B,X,Y)
{
    __shared__ float ldsA[WAVES_PER_BLOCK * LDS_WAVE];        // 36.9 KB

    const int lane = threadIdx.x & 31;
    const int wave = threadIdx.x >> 5;
    const int n  = lane & 15;                   // pixel index within tile (N of D')
    const int hi = lane >> 4;                   // half-wave select

    // block -> (b,x) row (uniform/SALU); wave -> 64-pixel y range
    const int b = blockIdx.x >> 9;
    const int x = blockIdx.x & (XDIM - 1);
    const int yBase = wave * (TILE * TILES_PER_WAVE);
    const long rowPix = ((long)b * XDIM + x) * YDIM;   // pixel index of (b,x,0)

    // ---- per-wave LDS / global staging offsets (tile-invariant)
    unsigned ldsWaveOff = (unsigned)(uintptr_t)
        (__attribute__((address_space(3))) float*)&ldsA[wave * LDS_WAVE];
    const int mrow = lane >> 3;                 // row within a 4-row group
    const int jseg = lane & 7;                  // 16B segment within a 128B row
    const unsigned lOffLane = (unsigned)(mrow * (LDS_ROW * 4) + jseg * 16);
    const unsigned gOffLane = (unsigned)(mrow * (KDIM * 4)  + jseg * 16);

    auto stage = [&](int t, int buf) {
        unsigned long long gbase = (unsigned long long)(uintptr_t)
            (kern + (rowPix + yBase + (long)t * TILE) * KDIM);
        unsigned lbase = ldsWaveOff + (unsigned)(buf * (LDS_TILE * 4));
#pragma unroll
        for (int i = 0; i < 4; ++i) {           // 4 async b128 ops = 2KB tile
            unsigned l_off = lbase + lOffLane + (unsigned)(i * 4 * (LDS_ROW * 4));
            unsigned g_off = gOffLane + (unsigned)(i * 4 * (KDIM * 4));
            asm volatile("global_load_async_to_lds_b128 %0, %1, %2"
                         :: "v"(l_off), "v"(g_off), "s"(gbase) : "memory");
        }
    };

    // ---- hoisted tile-invariant operands
    v2f nsf[8];                                 // nullspace^T A-fragments
#pragma unroll
    for (int kc = 0; kc < 8; ++kc) {
        const int kbase = kc * 4 + 2 * hi;
        nsf[kc].x = nullspace[(kbase + 0) * SDIM + n];
        nsf[kc].y = nullspace[(kbase + 1) * SDIM + n];
    }
    float bseed[8];                             // bias[s], s = v + 8*hi
#pragma unroll
    for (int v = 0; v < 8; ++v) bseed[v] = bias[v + 8 * hi];

    // stencil: offsets arange(4)-2+1 = {-1,0,1,2}; s = dxIdx*4 + dyIdx
    const float* sB = source + (long)b * (XDIM * YDIM);
    int rowOff[8], dyv[8];
#pragma unroll
    for (int v = 0; v < 8; ++v) {
        const int s = v + 8 * hi;
        dyv[v]    = (s & 3) - 1;
        rowOff[v] = ((x + ((s >> 2) - 1)) & (XDIM - 1)) * YDIM;
    }

    // ---- pipelined tile loop
    stage(0, 0);
#pragma unroll
    for (int t = 0; t < TILES_PER_WAVE; ++t) {
        if (t + 1 < TILES_PER_WAVE) {
            stage(t + 1, (t + 1) & 1);          // prefetch next tile into alt buffer
            asm volatile("s_wait_asynccnt 4" ::: "memory");   // oldest 4 (tile t) done
        } else {
            asm volatile("s_wait_asynccnt 0" ::: "memory");
        }

        const float* aRow = &ldsA[wave * LDS_WAVE + (t & 1) * LDS_TILE + n * LDS_ROW];
        v8f c;
#pragma unroll
        for (int v = 0; v < 8; ++v) c[v] = bseed[v];

#pragma unroll
        for (int kc = 0; kc < 8; ++kc) {
            const int kbase = kc * 4 + 2 * hi;
            v2f kf = *(const v2f*)(aRow + kbase);   // ds_load (merged to 2addr_b64)
            c = __builtin_amdgcn_wmma_f32_16x16x4_f32(false, nsf[kc], false, kf,
                                                      (short)0, c, false, false);
        }

        // lane owns pixel p = n of this tile, stencil slots s = v + 8*hi in regs
        const int y = yBase + t * TILE + n;
        float acc = 0.0f;
#pragma unroll
        for (int v = 0; v < 8; ++v) {
            const int yy = (y + dyv[v]) & (YDIM - 1);
            acc = fmaf(c[v], sB[rowOff[v] + yy], acc);
        }
        // combine the two stencil halves: single xor-16 swizzle
        int ai = __builtin_bit_cast(int, acc);
        float other = __builtin_bit_cast(float,
                          __builtin_amdgcn_ds_swizzle(ai, 0x401F));
        float tot = acc + other;
        if (hi == 0) out[rowPix + y] = tot;     // 16 contiguous b32 stores
    }
}

extern "C" void kernel_launch(void* const* d_in, const int* in_sizes, int n_in,
                              void* d_out, int out_size, void* d_ws, size_t ws_size,
                              hipStream_t stream) {
    const float* kern      = (const float*)d_in[0];  // (B,X,Y,32) f32
    const float* source    = (const float*)d_in[1];  // (B,X,Y)    f32
    const float* nullspace = (const float*)d_in[2];  // (32,16)    f32
    const float* bias      = (const float*)d_in[3];  // (16,)      f32
    float* out = (float*)d_out;                      // (B,X,Y)    f32

    // one block per (b,x) row: 8 waves x 4 tiles x 16 pixels = 512 = YDIM
    const int blocks = out_size / YDIM;              // 8192
    varcoef_wmma_kernel<<<blocks, 256, 0, stream>>>(kern, source, nullspace, bias, out);
}